// CTCLossLayer_6296422055981
// MI455X (gfx1250) — compile-verified
//
#include <hip/hip_runtime.h>
#include <math.h>

// Problem dimensions (fixed by the reference's setup_inputs).
#define CTC_B 256
#define CTC_T 512
#define CTC_C 128
#define CTC_U 128
// S = 2U+1 = 257 states: lane L owns states 8L..8L+7, plus state 256 on lane 31.

#define NEGF (-1e30f)
#define EPSF (1e-7f)

// ---- CDNA5 wait-counter helpers (builtin if present, else raw ISA) ----
#if __has_builtin(__builtin_amdgcn_s_wait_asynccnt)
#define WAIT_ASYNCCNT(n) __builtin_amdgcn_s_wait_asynccnt(n)
#else
#define WAIT_ASYNCCNT(n) asm volatile("s_wait_asynccnt %0" ::"i"(n) : "memory")
#endif

#if __has_builtin(__builtin_amdgcn_s_wait_dscnt)
#define WAIT_DSCNT(n) __builtin_amdgcn_s_wait_dscnt(n)
#else
#define WAIT_DSCNT(n) asm volatile("s_wait_dscnt %0" ::"i"(n) : "memory")
#endif

// 16-byte vector type matching the builtin's expected pointee type.
typedef int ctc_v4i __attribute__((vector_size(4 * sizeof(int))));
typedef __attribute__((address_space(1))) ctc_v4i* ctc_gvp;  // global
typedef __attribute__((address_space(3))) ctc_v4i* ctc_lvp;  // LDS

// ---- Async global -> LDS copy: 16 bytes per lane (512 B per wave = one prob row)
__device__ __forceinline__ void async_row16(const float* gsrc, float* ldst) {
#if __has_builtin(__builtin_amdgcn_global_load_async_to_lds_b128)
  __builtin_amdgcn_global_load_async_to_lds_b128(
      (ctc_gvp)(__attribute__((address_space(1))) void*)(void*)(gsrc),
      (ctc_lvp)(__attribute__((address_space(3))) void*)(void*)(ldst),
      /*offset=*/0, /*cpol=*/0);
#else
  // Fallback: synchronous vector copy (still correct; no ASYNCcnt traffic).
  *(float4*)ldst = *(const float4*)gsrc;
#endif
}

// shuffle-up by 1 within the wave32 (lane 0 result is masked by caller)
__device__ __forceinline__ float shfl_up1(float x) {
  int addr = (((int)threadIdx.x) - 1) << 2;  // byte address; HW uses bits [6:2]
  return __int_as_float(__builtin_amdgcn_ds_bpermute(addr, __float_as_int(x)));
}

// logaddexp, safe for the -1e30 "log(0)" sentinel (no NaNs, sentinel absorbs)
__device__ __forceinline__ float lae(float x, float y) {
  float m = fmaxf(x, y);
  float d = fminf(x, y) - m;  // <= 0
  return m + log1pf(__expf(d));
}

__global__ __launch_bounds__(32) void ctc_alpha_scan(
    const int* __restrict__ y_true,   // [B, U] int32, labels in [0, C-2]
    const float* __restrict__ y_pred, // [B, T, C] float32 probabilities
    float* __restrict__ out) {        // [B] (== [B,1] flattened)
  __shared__ float probs[2][CTC_C];  // double-buffered probability row

  const int b = blockIdx.x;
  const int lane = threadIdx.x;  // wave32: one wave per batch item
  const float* row_base = y_pred + (size_t)b * CTC_T * CTC_C;

  // Stage rows 0 and 1 asynchronously before doing anything else.
  async_row16(row_base + 0 * CTC_C + lane * 4, &probs[0][lane * 4]);
  async_row16(row_base + 1 * CTC_C + lane * 4, &probs[1][lane * 4]);

  // Labels this lane needs: states 8L+k, odd k -> label[4L + (k-1)/2];
  // the s-2 skip test for k=1 needs label[4L-1] from the neighbor block.
  const int* yt = y_true + b * CTC_U;
  const int li = lane * 4;
  const int labm1 = (lane > 0) ? yt[li - 1] : -1;
  const int lab0 = yt[li + 0];
  const int lab1 = yt[li + 1];
  const int lab2 = yt[li + 2];
  const int lab3 = yt[li + 3];

  // skip[s] for odd s >= 3: current label != label two slots back.
  // s = 1 (lane 0, k = 1) never skips. Even states never skip (blank).
  const bool sk1 = (lane > 0) && (lab0 != labm1);
  const bool sk3 = (lab1 != lab0);
  const bool sk5 = (lab2 != lab1);
  const bool sk7 = (lab3 != lab2);

  // alpha registers: a0..a7 = states 8L..8L+7, a8 = state 256 (lane 31 valid)
  float a0 = NEGF, a1 = NEGF, a2 = NEGF, a3 = NEGF, a4 = NEGF,
        a5 = NEGF, a6 = NEGF, a7 = NEGF, a8 = NEGF;

  for (int t = 0; t < CTC_T; ++t) {
    const int buf = t & 1;

    // Rows complete in order: after issuing row t+1, asynccnt<=1 means row t
    // has landed. Final iteration may have itself as the single outstanding
    // op, so drain fully there.
    if (t + 1 < CTC_T) {
      WAIT_ASYNCCNT(1);
    } else {
      WAIT_ASYNCCNT(0);
    }

    // Gather the 5 probabilities this lane needs (LDS, random-ish banks).
    float pb = probs[buf][CTC_C - 1];  // blank
    float p1 = probs[buf][lab0];
    float p3 = probs[buf][lab1];
    float p5 = probs[buf][lab2];
    float p7 = probs[buf][lab3];
    float lpb = logf(pb + EPSF);
    float lp1 = logf(p1 + EPSF);
    float lp3 = logf(p3 + EPSF);
    float lp5 = logf(p5 + EPSF);
    float lp7 = logf(p7 + EPSF);

    // Make sure the DS reads of this buffer retired before the async engine
    // overwrites it, then kick off row t+2 so it overlaps the VALU chain.
    WAIT_DSCNT(0);
    if (t + 2 < CTC_T) {
      async_row16(row_base + (size_t)(t + 2) * CTC_C + lane * 4,
                  &probs[buf][lane * 4]);
    }

    if (t == 0) {
      // alpha0: only states 0 (blank) and 1 (first label) are live.
      a0 = (lane == 0) ? lpb : NEGF;
      a1 = (lane == 0) ? lp1 : NEGF;
      // a2..a8 stay NEGF
    } else {
      // Only cross-lane dependency: state 8L-1 = neighbor's a7
      // (feeds n0's s-1 term and n1's s-2 skip term).
      float pm1 = shfl_up1(a7);
      pm1 = (lane == 0) ? NEGF : pm1;

      float n0 = lae(a0, pm1) + lpb;
      float n1 = lae(lae(a1, a0), sk1 ? pm1 : NEGF) + lp1;
      float n2 = lae(a2, a1) + lpb;
      float n3 = lae(lae(a3, a2), sk3 ? a1 : NEGF) + lp3;
      float n4 = lae(a4, a3) + lpb;
      float n5 = lae(lae(a5, a4), sk5 ? a3 : NEGF) + lp5;
      float n6 = lae(a6, a5) + lpb;
      float n7 = lae(lae(a7, a6), sk7 ? a5 : NEGF) + lp7;
      float n8 = lae(a8, a7) + lpb;  // state 256 (valid on lane 31 only)

      a0 = n0; a1 = n1; a2 = n2; a3 = n3; a4 = n4;
      a5 = n5; a6 = n6; a7 = n7; a8 = n8;
    }
  }

  // loss = -logaddexp(alpha[S-1], alpha[S-2]) = -lae(state256, state255)
  if (lane == 31) out[b] = -lae(a8, a7);
}

extern "C" void kernel_launch(void* const* d_in, const int* in_sizes, int n_in,
                              void* d_out, int out_size, void* d_ws, size_t ws_size,
                              hipStream_t stream) {
  (void)in_sizes; (void)n_in; (void)out_size; (void)d_ws; (void)ws_size;
  const int* y_true = (const int*)d_in[0];      // [256,128] int32
  const float* y_pred = (const float*)d_in[1];  // [256,512,128] float32
  float* out = (float*)d_out;                   // [256,1] float32

  ctc_alpha_scan<<<dim3(CTC_B), dim3(32), 0, stream>>>(y_true, y_pred, out);
}